// GCN_27307402068240
// MI455X (gfx1250) — compile-verified
//
#include <hip/hip_runtime.h>
#include <hip/hip_bf16.h>

#define N_NODES   50000
#define N_EDGES   800000
#define DIM       128
#define N_CLASSES 10
#define N_GRAPHS  64
#define NEG_SLOPE 0.2f

typedef __attribute__((ext_vector_type(2))) float v2f;
typedef __attribute__((ext_vector_type(8))) float v8f;

// ---------------------------------------------------------------- utilities
__global__ void zero_kernel(float* __restrict__ p, int n) {
    int i = blockIdx.x * blockDim.x + threadIdx.x;
    if (i < n) p[i] = 0.0f;
}

// deg_out[src[e]] += 1 ; deg_in[dst[e]] += 1
__global__ void degree_kernel(const int* __restrict__ src, const int* __restrict__ dst,
                              float* __restrict__ degOut, float* __restrict__ degIn) {
    int e = blockIdx.x * blockDim.x + threadIdx.x;
    if (e < N_EDGES) {
        atomicAdd(&degOut[src[e]], 1.0f);
        atomicAdd(&degIn[dst[e]], 1.0f);
    }
}

// norm = rsqrt(max(deg, 1))
__global__ void norm_kernel(const float* __restrict__ deg, float* __restrict__ norm, int n) {
    int i = blockIdx.x * blockDim.x + threadIdx.x;
    if (i < n) norm[i] = rsqrtf(fmaxf(deg[i], 1.0f));
}

// ------------------------------------------------------------- WMMA GEMM
// Y[row0..row0+16, 0..128] = pre(Ain) @ W   with per-block 16-row tile.
// pre(a,i,j) = actEn ? leaky_relu(a*normDst[i] + bias[j]) * normSrc[i]
//                    : a * normSrc[i]
// Block = 256 threads = 8 waves; wave w owns output columns [16w, 16w+16).
__global__ __launch_bounds__(256) void gemm_kernel(
        const float* __restrict__ Ain, const float* __restrict__ W,
        const float* __restrict__ normSrc, const float* __restrict__ normDst,
        const float* __restrict__ bias, int actEn, float* __restrict__ Y) {
    __shared__ float lAs[16][DIM + 1];   // +1 pad: kills 16-way bank conflict

    const int row0 = blockIdx.x * 16;

    // Stage A tile into LDS with fused pre-activation / norm scaling.
    for (int t = threadIdx.x; t < 16 * DIM; t += 256) {
        int r  = t >> 7;         // /128
        int c  = t & (DIM - 1);
        int gi = row0 + r;
        float a = Ain[(size_t)gi * DIM + c];
        if (actEn) {
            a = a * normDst[gi] + bias[c];
            a = (a > 0.0f) ? a : a * NEG_SLOPE;
        }
        a *= normSrc[gi];
        lAs[r][c] = a;
    }
    __syncthreads();

    const int lane = threadIdx.x & 31;
    const int hl   = lane >> 4;       // half-wave: 0 or 1
    const int mn   = lane & 15;       // M (for A) / N (for B,C,D)
    const int colB = (threadIdx.x >> 5) * 16;   // wave's N tile

    v8f c = {};
#pragma unroll
    for (int k0 = 0; k0 < DIM; k0 += 4) {
        const int ka = k0 + hl * 2;
        v2f a, b;
        // A 16x4 f32 frag: VGPR v, lane l -> M=l%16, K=(l/16)*2+v
        a.x = lAs[mn][ka + 0];
        a.y = lAs[mn][ka + 1];
        // B 4x16 f32 frag (mirror): N=l%16, K=(l/16)*2+v
        b.x = W[(size_t)(ka + 0) * DIM + colB + mn];
        b.y = W[(size_t)(ka + 1) * DIM + colB + mn];
        c = __builtin_amdgcn_wmma_f32_16x16x4_f32(
                /*neg_a=*/false, a, /*neg_b=*/false, b,
                /*c_mod=*/(short)0, c, /*reuse_a=*/false, /*reuse_b=*/false);
    }

    // C/D 16x16 f32: VGPR r, lane l -> M = r + 8*(l/16), N = l%16
#pragma unroll
    for (int r = 0; r < 8; ++r) {
        int grow = row0 + r + hl * 8;
        Y[(size_t)grow * DIM + colB + mn] = c[r];
    }
}

// ---------------------------------------------------- edge scatter-add
// agg[dst[e], :] += Y[src[e], :]   (32 lanes/edge, float4 per lane)
__global__ __launch_bounds__(256) void aggregate_kernel(
        const float* __restrict__ Y, const int* __restrict__ src,
        const int* __restrict__ dst, float* __restrict__ agg) {
    int idx = blockIdx.x * blockDim.x + threadIdx.x;
    int e   = idx >> 5;
    int c4  = (idx & 31) << 2;
    if (e < N_EDGES) {
        int s = src[e], d = dst[e];
        float4 v = *reinterpret_cast<const float4*>(Y + (size_t)s * DIM + c4);
        float* o = agg + (size_t)d * DIM + c4;
        atomicAdd(o + 0, v.x);
        atomicAdd(o + 1, v.y);
        atomicAdd(o + 2, v.z);
        atomicAdd(o + 3, v.w);
    }
}

// -------------------------------------------- graph mean-pool (pass 1)
// h3 = leaky_relu(agg*normDst + b3); sums[g,:] += h3[i,:]; counts[g] += 1
__global__ __launch_bounds__(128) void pool_kernel(
        const float* __restrict__ agg, const float* __restrict__ normDst,
        const float* __restrict__ bias, const int* __restrict__ gid,
        float* __restrict__ sums, float* __restrict__ counts) {
    int i = blockIdx.x;      // node
    int j = threadIdx.x;     // feature
    float a = agg[(size_t)i * DIM + j] * normDst[i] + bias[j];
    a = (a > 0.0f) ? a : a * NEG_SLOPE;
    int g = gid[i];
    atomicAdd(&sums[g * DIM + j], a);
    if (j == 0) atomicAdd(&counts[g], 1.0f);
}

// ------------------------------------------------------ final dense head
// out[g,c] = (sums[g,:]/max(counts[g],1)) @ Wd[:,c] + bd[c]
__global__ void dense_kernel(const float* __restrict__ sums,
                             const float* __restrict__ counts,
                             const float* __restrict__ Wd,
                             const float* __restrict__ bd,
                             float* __restrict__ out) {
    int t = blockIdx.x * blockDim.x + threadIdx.x;
    if (t < N_GRAPHS * N_CLASSES) {
        int g = t / N_CLASSES, c = t % N_CLASSES;
        float inv = 1.0f / fmaxf(counts[g], 1.0f);
        float acc = bd[c];
#pragma unroll 4
        for (int j = 0; j < DIM; ++j)
            acc += sums[g * DIM + j] * inv * Wd[j * N_CLASSES + c];
        out[t] = acc;
    }
}

// -------------------------------------------------------------- launcher
extern "C" void kernel_launch(void* const* d_in, const int* in_sizes, int n_in,
                              void* d_out, int out_size, void* d_ws, size_t ws_size,
                              hipStream_t stream) {
    const float* x   = (const float*)d_in[0];
    const int*   ei  = (const int*)d_in[1];
    const int*   gid = (const int*)d_in[2];
    const float* W1  = (const float*)d_in[3];
    const float* b1  = (const float*)d_in[4];
    const float* W2  = (const float*)d_in[5];
    const float* b2  = (const float*)d_in[6];
    const float* W3  = (const float*)d_in[7];
    const float* b3  = (const float*)d_in[8];
    const float* Wd  = (const float*)d_in[9];
    const float* bd  = (const float*)d_in[10];
    float* out = (float*)d_out;

    const int* src = ei;
    const int* dst = ei + N_EDGES;

    // Workspace carve (floats): degOut|degIn|normS|normD|counts|sums|Y|agg
    float* ws     = (float*)d_ws;
    float* degOut = ws;
    float* degIn  = degOut + N_NODES;
    float* normS  = degIn + N_NODES;
    float* normD  = normS + N_NODES;
    float* counts = normD + N_NODES;
    float* sums   = counts + N_GRAPHS;
    float* Y      = sums + (size_t)N_GRAPHS * DIM;           // 16B-aligned offset
    float* agg    = Y + (size_t)N_NODES * DIM;
    (void)ws_size; (void)n_in; (void)in_sizes; (void)out_size;

    const int B = 256;
    // Degrees + norms (deg/norm arrays are contiguous pairs -> one pass each)
    zero_kernel<<<(2 * N_NODES + B - 1) / B, B, 0, stream>>>(degOut, 2 * N_NODES);
    degree_kernel<<<(N_EDGES + B - 1) / B, B, 0, stream>>>(src, dst, degOut, degIn);
    norm_kernel<<<(2 * N_NODES + B - 1) / B, B, 0, stream>>>(degOut, normS, 2 * N_NODES);

    const int gemmGrid = N_NODES / 16;            // 3125 (exact)
    const int aggGrid  = (N_EDGES * 32) / B;      // 100000 (exact)
    const int zeroFeat = (N_NODES * DIM + B - 1) / B;

    // Layer 1: A = x * normS
    gemm_kernel<<<gemmGrid, B, 0, stream>>>(x, W1, normS, nullptr, nullptr, 0, Y);
    zero_kernel<<<zeroFeat, B, 0, stream>>>(agg, N_NODES * DIM);
    aggregate_kernel<<<aggGrid, B, 0, stream>>>(Y, src, dst, agg);

    // Layer 2: A = leaky(agg*normD + b1) * normS
    gemm_kernel<<<gemmGrid, B, 0, stream>>>(agg, W2, normS, normD, b1, 1, Y);
    zero_kernel<<<zeroFeat, B, 0, stream>>>(agg, N_NODES * DIM);
    aggregate_kernel<<<aggGrid, B, 0, stream>>>(Y, src, dst, agg);

    // Layer 3: A = leaky(agg*normD + b2) * normS
    gemm_kernel<<<gemmGrid, B, 0, stream>>>(agg, W3, normS, normD, b2, 1, Y);
    zero_kernel<<<zeroFeat, B, 0, stream>>>(agg, N_NODES * DIM);
    aggregate_kernel<<<aggGrid, B, 0, stream>>>(Y, src, dst, agg);

    // Mean pool + dense head
    zero_kernel<<<(N_GRAPHS * (DIM + 1) + B - 1) / B, B, 0, stream>>>(
        counts, N_GRAPHS * (DIM + 1));
    pool_kernel<<<N_NODES, DIM, 0, stream>>>(agg, normD, b3, gid, sums, counts);
    dense_kernel<<<(N_GRAPHS * N_CLASSES + B - 1) / B, B, 0, stream>>>(
        sums, counts, Wd, bd, out);
}